// SpodNet_61692910239920
// MI455X (gfx1250) — compile-verified
//
#include <hip/hip_runtime.h>

// ---------------------------------------------------------------------------
// SpodNet one-pass column-sweep for B=2048 batches of 32x32 (P=32) matrices.
//
// Mapping: 1 wave32 = 1 batch (lane = row). 8 waves (256 thr) per workgroup.
// Theta/W live in LDS (row stride 33 floats -> conflict-free column gathers).
// Per-batch matvecs: unrolled register arrays + v_readlane broadcasts.
// The batch-shared L-transform (theta_12 @ L^T) is batched across the 8
// waves of the workgroup and computed by wave 0 with V_WMMA_F32_16X16X4_F32
// (the only f32 WMMA shape on CDNA5) -- f32 is required for the numerically
// sensitive inverse-maintenance updates.
// ---------------------------------------------------------------------------

typedef float v2f __attribute__((ext_vector_type(2)));
typedef float v8f __attribute__((ext_vector_type(8)));

#define PP        32
#define S         33          // LDS row stride (33*i mod 64 -> 32 distinct banks)
#define WAVES     8           // batches per workgroup
#define THREADS   (WAVES * 32)
#define MAT       (PP * S)    // 1056 floats per matrix

// LDS float offsets
#define OFF_TH    0
#define OFF_W     (WAVES * MAT)                 //  8448
#define OFF_LRAW  (2 * WAVES * MAT)             // 16896  (31*31 raw L)
#define OFF_LT    (OFF_LRAW + 31 * 31)          // 17857  (32x32 eff. L^T, stride S)
#define OFF_T12   (OFF_LT + PP * S)             // 18913  (16 x S)
#define OFF_TN    (OFF_T12 + 16 * S)            // 19441  (16 x S)
#define SMEM_FLTS (OFF_TN + 16 * S)             // 19969 floats = 79876 B

#if defined(__has_builtin)
#if __has_builtin(__builtin_amdgcn_wmma_f32_16x16x4_f32)
#define HAVE_WMMA_F32X4 1
#endif
#endif

__global__ __launch_bounds__(THREADS)
void spodnet_pass_kernel(const float* __restrict__ ThetaG,
                         const float* __restrict__ WG,
                         const float* __restrict__ LG,
                         float* __restrict__ outG)
{
    extern __shared__ float sm[];
    float* Th  = sm + OFF_TH;     // per-wave: Th  + w*MAT
    float* Wm  = sm + OFF_W;      // per-wave: Wm  + w*MAT
    float* Lr  = sm + OFF_LRAW;   // raw L, 31x31 row-major
    float* Lt  = sm + OFF_LT;     // per-column effective L^T (K x N), stride S
    float* T12 = sm + OFF_T12;    // 16 x S staging (rows 8..15 stay zero)
    float* TN  = sm + OFF_TN;     // 16 x S WMMA result

    const int tid  = threadIdx.x;
    const int lane = tid & 31;
    const int w    = tid >> 5;
    const int b    = blockIdx.x * WAVES + w;

    float* th = Th + w * MAT;
    float* wm = Wm + w * MAT;

    // ---- init: coalesced global->LDS copy of this wave's Theta and W ----
    {
        const float* gT = ThetaG + (size_t)b * (PP * PP);
        const float* gW = WG     + (size_t)b * (PP * PP);
        #pragma unroll 4
        for (int e = lane; e < PP * PP; e += 32) {
            int r = e >> 5, c = e & 31;
            th[r * S + c] = gT[e];
            wm[r * S + c] = gW[e];
        }
    }
    for (int e = tid; e < 31 * 31; e += THREADS) Lr[e] = LG[e];
    for (int e = tid; e < 8 * S;  e += THREADS) T12[8 * S + e] = 0.0f; // pad rows

    // ---- sequential column sweep (scan dependency) ----
    for (int col = 0; col < PP; ++col) {
        const bool diag = (lane == col);

        float t12 = th[lane * S + col];
        float w12 = wm[lane * S + col];
        if (diag) { t12 = 0.0f; w12 = 0.0f; }   // zero-pad removed index
        const float w22  = wm[col * S + col];
        const float th22 = th[col * S + col];
        const float rw22 = 1.0f / w22;

        T12[w * S + lane] = t12;

        // Effective L^T for this col: Lt[k][n] = L[pos(n)][pos(k)], pos shifts at col
        for (int e = tid; e < PP * PP; e += THREADS) {
            int k = e >> 5, n = e & 31;
            float v = 0.0f;
            if (k != col && n != col) {
                int pk = k - (k > col);
                int pn = n - (n > col);
                v = Lr[pn * 31 + pk];
            }
            Lt[k * S + n] = v;
        }

        // M row (registers) + u = M @ theta_12, overlapped before the barrier
        float M[PP], Wrow[PP];
        float u = 0.0f;
        #pragma unroll
        for (int j = 0; j < PP; ++j) {
            float wj  = __shfl(w12, j, 32);       // v_readlane broadcasts
            float tj  = __shfl(t12, j, 32);
            float wij = wm[lane * S + j];
            Wrow[j] = wij;
            float m = fmaf(-w12 * wj, rw22, wij); // W11 - w12 w12^T / w22
            m = (diag || j == col) ? 0.0f : m;
            M[j] = m;
            u = fmaf(m, tj, u);
        }
        // gy = theta22 - theta12^T M theta12  (wave32 butterfly reduce)
        float red = t12 * u;
        #pragma unroll
        for (int off = 16; off; off >>= 1) red += __shfl_xor(red, off, 32);
        const float gy = th22 - red;

        __syncthreads();   // T12 + Lt ready for the cooperative transform

        if (w == 0) {
            // TN(16x32) = T12(16x32) @ Lt(32x32) for all 8 batches at once.
#ifdef HAVE_WMMA_F32X4
            const int half = (lane < 16) ? 0 : 1;
            const int r    = lane & 15;
            #pragma unroll
            for (int nt = 0; nt < 2; ++nt) {
                const int n0 = nt * 16 + r;       // B/D column this lane owns
                v8f acc = {};
                #pragma unroll
                for (int k4 = 0; k4 < 8; ++k4) {  // K = 32 in steps of 4
                    v2f a, bv;
                    // A 16x4 f32 layout: lanes 0-15 K={0,1}, lanes 16-31 K={2,3}
                    a.x  = T12[r * S + 4 * k4 + (half ? 2 : 0)];
                    a.y  = T12[r * S + 4 * k4 + (half ? 3 : 1)];
                    // B 4x16 f32 layout (mirrors A): K rows striped across halves
                    bv.x = Lt[(4 * k4 + (half ? 2 : 0)) * S + n0];
                    bv.y = Lt[(4 * k4 + (half ? 3 : 1)) * S + n0];
                    acc = __builtin_amdgcn_wmma_f32_16x16x4_f32(
                        false, a, false, bv, (short)0, acc, false, false);
                }
                // D 16x16 f32: VGPR rr -> (M=rr, lanes 0-15) / (M=rr+8, lanes 16-31)
                #pragma unroll
                for (int rr = 0; rr < 8; ++rr)
                    TN[(rr + (half ? 8 : 0)) * S + n0] = acc[rr];
            }
#else
            for (int e = lane; e < WAVES * PP; e += 32) {
                int r = e >> 5, n = e & 31;
                float s = 0.0f;
                for (int k = 0; k < PP; ++k)
                    s = fmaf(T12[r * S + k], Lt[k * S + n], s);
                TN[r * S + n] = s;
            }
#endif
        }
        __syncthreads();   // TN ready

        const float tn = TN[w * S + lane];       // theta_12_next (0 at col)

        // q = M @ tn
        float q = 0.0f;
        #pragma unroll
        for (int j = 0; j < PP; ++j) {
            float tnj = __shfl(tn, j, 32);
            q = fmaf(M[j], tnj, q);
        }
        float sred = tn * q;                     // schur = <tn, q>
        #pragma unroll
        for (int off = 16; off; off >>= 1) sred += __shfl_xor(sred, off, 32);

        const float th22n = gy + sred;
        const float w22n  = 1.0f / gy;
        const float w12n  = -w22n * q;           // == 0 at lane==col

        // Theta row/col/diag update
        const float tval = diag ? th22n : tn;
        th[lane * S + col] = tval;
        th[col * S + lane] = tval;

        // W update: 2*W11 - (M + gy * w12n w12n^T), with branch-free
        // corrections folding in W12_new / w22_new (w12n[col]==0 makes it exact)
        #pragma unroll
        for (int j = 0; j < PP; ++j) {
            float bj = __shfl(w12n, j, 32);
            float v  = 2.0f * Wrow[j] - M[j] - gy * w12n * bj;
            if (j == col)         v -= w12n;     // -> W12_new[i] = 2 w12 - w12n
            if (diag)             v -= bj;       // -> row col symmetric
            if (diag && j == col) v -= w22n;     // -> w22_new = 2 w22 - w22n
            wm[lane * S + j] = v;
        }
        // No extra barrier needed: next column's shared writes (T12/Lt) only
        // race with wave0's WMMA of THIS column, which completed before the
        // second barrier above.
    }

    // ---- coalesced LDS -> global writeback of Theta ----
    {
        float* gO = outG + (size_t)b * (PP * PP);
        #pragma unroll 4
        for (int e = lane; e < PP * PP; e += 32) {
            int r = e >> 5, c = e & 31;
            gO[e] = th[r * S + c];
        }
    }
}

extern "C" void kernel_launch(void* const* d_in, const int* in_sizes, int n_in,
                              void* d_out, int out_size, void* d_ws, size_t ws_size,
                              hipStream_t stream)
{
    (void)in_sizes; (void)n_in; (void)out_size; (void)d_ws; (void)ws_size;
    const float* Theta = (const float*)d_in[0];   // 2048*32*32
    const float* W     = (const float*)d_in[1];   // 2048*32*32
    const float* L     = (const float*)d_in[2];   // 31*31
    float* out         = (float*)d_out;           // 2048*32*32

    const int B = 2048;
    const int blocks = B / WAVES;                 // 256 workgroups
    const size_t smem = (size_t)SMEM_FLTS * sizeof(float);  // ~78 KB (<320 KB WGP)

    spodnet_pass_kernel<<<dim3(blocks), dim3(THREADS), smem, stream>>>(
        Theta, W, L, out);
}